// TransducerJoint_16338055594593
// MI455X (gfx1250) — compile-verified
//
#include <hip/hip_runtime.h>
#include <hip/hip_bf16.h>
#include <math.h>

typedef float v2f __attribute__((ext_vector_type(2)));
typedef float v8f __attribute__((ext_vector_type(8)));

#define B_  4
#define T_  200
#define U_  50
#define E_  512
#define J_  512
#define V_  4096
#define M_JOINT (B_*T_*U_)   // 40000, divisible by 64

static __device__ __forceinline__ v8f wmma_f32(v2f a, v2f b, v8f c) {
  // D = A(16x4,f32) * B(4x16,f32) + C(16x16,f32)
  return __builtin_amdgcn_wmma_f32_16x16x4_f32(false, a, false, b, (short)0, c, false, false);
}

// ---------------------------------------------------------------------------
// out_w [V,J] row-major  ->  w_t [J,V] row-major (coalesced B-fragment loads)
// ---------------------------------------------------------------------------
__global__ void k_transpose_w(const float* __restrict__ w, float* __restrict__ wt) {
  __shared__ float tile[32][33];
  const int j0 = blockIdx.y * 32;
  const int v0 = blockIdx.x * 32;
  const int tx = threadIdx.x, ty = threadIdx.y;   // 32 x 8
#pragma unroll
  for (int i = 0; i < 4; ++i) {
    int v = v0 + ty + i * 8;
    tile[ty + i * 8][tx] = w[(size_t)v * J_ + (j0 + tx)];
  }
  __syncthreads();
#pragma unroll
  for (int i = 0; i < 4; ++i) {
    int j = j0 + ty + i * 8;
    wt[(size_t)j * V_ + (v0 + tx)] = tile[tx][ty + i * 8];
  }
}

// ---------------------------------------------------------------------------
// Y[m, j] = sum_k X[m,k] * W[j,k] + bias[j]     (torch Linear)
// One wave per 16x16 output tile, K stepped by 4 with V_WMMA_F32_16X16X4_F32.
//   A 16x4 layout : lane = M (l&15), K pair selected by half-wave (l>>4)
//   B  4x16 layout: lane = N (l&15), K pair selected by half-wave
// ---------------------------------------------------------------------------
__global__ void k_proj(const float* __restrict__ X, const float* __restrict__ W,
                       const float* __restrict__ bias, float* __restrict__ Y,
                       int Mreal, int Mtiles) {
  const int gw = blockIdx.x * (blockDim.x >> 5) + (threadIdx.x >> 5);
  const int total = Mtiles * (J_ / 16);
  if (gw >= total) return;                 // uniform per wave: EXEC stays all-1s
  const int jt = gw & (J_ / 16 - 1);
  const int mt = gw >> 5;
  const int l = threadIdx.x & 31, half = l >> 4, ln = l & 15;

  int arow = mt * 16 + ln;
  if (arow > Mreal - 1) arow = Mreal - 1;  // clamp reads; padded rows store garbage into ws
  const float* ap = X + (size_t)arow * E_;
  const float* bp = W + (size_t)(jt * 16 + ln) * E_;

  v8f acc = {};
#pragma unroll 4
  for (int k0 = 0; k0 < E_; k0 += 4) {
    const int k = k0 + half * 2;
    v2f a = *(const v2f*)(ap + k);
    v2f b = *(const v2f*)(bp + k);
    acc = wmma_f32(a, b, acc);
  }
  const float bs = bias[jt * 16 + ln];
#pragma unroll
  for (int i = 0; i < 8; ++i) {
    const int m = mt * 16 + i + half * 8;
    Y[(size_t)m * J_ + (jt * 16 + ln)] = acc[i] + bs;
  }
}

// ---------------------------------------------------------------------------
// Fused: A[m,k] = tanh(enc_act[bt,k] + pred_act[b*U+u,k]) staged in LDS once
// per 64-row block, then [64 x 512] x [512 x 4096] with fp32 WMMA.
// Per wave: 4 M-subtiles x 2 N-tiles -> 8 accumulators, B frag reused 4x,
// A frag reused 2x. LDS row stride 516 (mod 64 == 4) -> bank-conflict-free.
// ---------------------------------------------------------------------------
__device__ __forceinline__ void store_pair(float* __restrict__ out, int mbase, int half,
                                           int col0, int col1, const v8f& A0, const v8f& A1,
                                           float bs0, float bs1) {
#pragma unroll
  for (int i = 0; i < 8; ++i) {
    const size_t m = (size_t)(mbase + i + half * 8);
    out[m * (size_t)V_ + col0] = A0[i] + bs0;
    out[m * (size_t)V_ + col1] = A1[i] + bs1;
  }
}

#define LDSTRIDE 516

__global__ void __launch_bounds__(256)
k_joint(const float* __restrict__ enc, const float* __restrict__ pred,
        const float* __restrict__ wt, const float* __restrict__ ob,
        float* __restrict__ out) {
  __shared__ float At[64 * LDSTRIDE];      // 132,096 B of the WGP's 320 KB LDS
  __shared__ int rEnc[64];
  __shared__ int rPred[64];

  const int m0 = blockIdx.x * 64;
  const int t = threadIdx.x;

  if (t < 64) {
    const int m = m0 + t;
    const int u = m % U_;
    const int bt = m / U_;
    const int b = bt / T_;
    rEnc[t] = bt;
    rPred[t] = b * U_ + u;
  }
  __syncthreads();

  // Phase 1: tanh(enc + pred) into LDS. 64*512 elems / 256 threads = 128 each.
  for (int i = t; i < 64 * J_; i += 256) {
    const int r = i >> 9;
    const int c = i & (J_ - 1);
    At[r * LDSTRIDE + c] =
        tanhf(enc[(size_t)rEnc[r] * J_ + c] + pred[(size_t)rPred[r] * J_ + c]);
  }
  __syncthreads();

  // Phase 2: 8 waves, each covers 2 N-tiles x 16 outer iterations = 256 tiles.
  const int w = t >> 5, l = t & 31, half = l >> 4, ln = l & 15;

  for (int outer = 0; outer < 16; ++outer) {
    const int nt = outer * 16 + w * 2;
    const int n0 = nt * 16;
    const float* b0p = wt + n0 + ln;
    const float* b1p = wt + n0 + 16 + ln;

    v8f a00 = {}, a01 = {}, a10 = {}, a11 = {}, a20 = {}, a21 = {}, a30 = {}, a31 = {};

#pragma unroll 2
    for (int k0 = 0; k0 < J_; k0 += 4) {
      const int k = k0 + half * 2;
      v2f b0, b1;
      b0.x = b0p[(size_t)k * V_];
      b0.y = b0p[(size_t)(k + 1) * V_];
      b1.x = b1p[(size_t)k * V_];
      b1.y = b1p[(size_t)(k + 1) * V_];
      v2f fa0 = *(const v2f*)(At + (0 * 16 + ln) * LDSTRIDE + k);
      v2f fa1 = *(const v2f*)(At + (1 * 16 + ln) * LDSTRIDE + k);
      v2f fa2 = *(const v2f*)(At + (2 * 16 + ln) * LDSTRIDE + k);
      v2f fa3 = *(const v2f*)(At + (3 * 16 + ln) * LDSTRIDE + k);
      a00 = wmma_f32(fa0, b0, a00);  a01 = wmma_f32(fa0, b1, a01);
      a10 = wmma_f32(fa1, b0, a10);  a11 = wmma_f32(fa1, b1, a11);
      a20 = wmma_f32(fa2, b0, a20);  a21 = wmma_f32(fa2, b1, a21);
      a30 = wmma_f32(fa3, b0, a30);  a31 = wmma_f32(fa3, b1, a31);
    }

    const float bs0 = ob[n0 + ln];
    const float bs1 = ob[n0 + 16 + ln];
    store_pair(out, m0 + 0,  half, n0 + ln, n0 + 16 + ln, a00, a01, bs0, bs1);
    store_pair(out, m0 + 16, half, n0 + ln, n0 + 16 + ln, a10, a11, bs0, bs1);
    store_pair(out, m0 + 32, half, n0 + ln, n0 + 16 + ln, a20, a21, bs0, bs1);
    store_pair(out, m0 + 48, half, n0 + ln, n0 + 16 + ln, a30, a31, bs0, bs1);
  }
}

// ---------------------------------------------------------------------------
extern "C" void kernel_launch(void* const* d_in, const int* in_sizes, int n_in,
                              void* d_out, int out_size, void* d_ws, size_t ws_size,
                              hipStream_t stream) {
  const float* enc_out  = (const float*)d_in[0];   // [4,200,512]
  const float* pred_out = (const float*)d_in[1];   // [4,50,512]
  const float* enc_w    = (const float*)d_in[2];   // [512,512]
  const float* enc_b    = (const float*)d_in[3];   // [512]
  const float* pred_w   = (const float*)d_in[4];   // [512,512]
  const float* pred_b   = (const float*)d_in[5];   // [512]
  const float* out_w    = (const float*)d_in[6];   // [4096,512]
  const float* out_b    = (const float*)d_in[7];   // [4096]
  float* out = (float*)d_out;                      // [4,200,50,4096]

  float* ws       = (float*)d_ws;
  float* enc_act  = ws;                            // 800*512  floats
  float* pred_act = enc_act + 800 * J_;            // 208*512  floats (padded)
  float* w_t      = pred_act + 208 * J_;           // 512*4096 floats
  // total workspace: ~10.5 MB

  // 1) transpose out_w -> w_t [J,V]
  dim3 tb(32, 8);
  dim3 tg(V_ / 32, J_ / 32);
  k_transpose_w<<<tg, tb, 0, stream>>>(out_w, w_t);

  // 2) projections (fp32 WMMA): enc 50 M-tiles, pred 13 M-tiles (padded to 208)
  k_proj<<<(50 * 32 + 7) / 8, 256, 0, stream>>>(enc_out, enc_w, enc_b, enc_act, 800, 50);
  k_proj<<<(13 * 32 + 7) / 8, 256, 0, stream>>>(pred_out, pred_w, pred_b, pred_act, 200, 13);

  // 3) fused tanh-join + output GEMM + bias
  k_joint<<<M_JOINT / 64, 256, 0, stream>>>(enc_act, pred_act, w_t, out_b, out);
}